// BPS_7318624272411
// MI455X (gfx1250) — compile-verified
//
#include <hip/hip_runtime.h>
#include <math.h>

typedef __attribute__((ext_vector_type(2))) float v2f;
typedef __attribute__((ext_vector_type(8))) float v8f;

#define L_SYM 65536
#define MTA   64
#define NFW   64
#define FLEN  (2*NFW + 1)
#define TLB   64                       // time steps per block in conv kernel
#define PI_F  3.14159265358979323846f

#if defined(__has_builtin)
#if __has_builtin(__builtin_amdgcn_global_load_async_to_lds_b128)
#define HAVE_ASYNC_LDS 1
#endif
#endif

#ifdef HAVE_ASYNC_LDS
// Builtin prototype (from hipcc diagnostic): param0 is v4i in AS1 (global),
// param1 is v4i in AS3 (LDS), then constant int offset + cpol.
typedef __attribute__((__vector_size__(16))) int v4i;
typedef __attribute__((address_space(1))) v4i gv4i;
typedef __attribute__((address_space(3))) v4i lv4i;
#endif

// ---------------------------------------------------------------------------
// Kernel 1: dist[l][m] = |x|^2 + min_s( |s|^2 - 2*Re(EE * conj(s)) )
// via V_WMMA_F32_16X16X4_F32:  D = A(16x4) * B(4x16) + C
//   A rows  = 16 test angles of this l:  [EEre, EEim, 0, 0]
//   B cols  = 16 constellation points:   [-2*sre; -2*sim; 0; 0]
//   C       = |s|^2 broadcast down each column
// ISA layout: lanes 0-15 hold K0/K1 in v0/v1; lanes 16-31 hold K2/K3 -> zeros.
// ---------------------------------------------------------------------------
__global__ __launch_bounds__(256) void bps_dist_kernel(
    const float* __restrict__ xr_g, const float* __restrict__ xi_g,
    const float* __restrict__ sr_g, const float* __restrict__ si_g,
    float* __restrict__ dist)
{
  const int lane = threadIdx.x & 31;
  const int wave = threadIdx.x >> 5;
  const int l    = blockIdx.x * 8 + wave;
  const int n    = lane & 15;
  const bool lo  = lane < 16;

  const float xr = xr_g[l];
  const float xi = xi_g[l];
  const float px = xr * xr + xi * xi;

  v2f B[4];
  v8f C[4];
#pragma unroll
  for (int ct = 0; ct < 4; ++ct) {
    float sr  = sr_g[ct * 16 + n];
    float si  = si_g[ct * 16 + n];
    float ssq = sr * sr + si * si;
    v2f b;
    b.x = lo ? (-2.0f * sr) : 0.0f;
    b.y = lo ? (-2.0f * si) : 0.0f;
    B[ct] = b;
    v8f c;
#pragma unroll
    for (int r = 0; r < 8; ++r) c[r] = ssq;
    C[ct] = c;
  }

#pragma unroll
  for (int t = 0; t < 4; ++t) {
    float theta = -0.25f * PI_F + (float)(t * 16 + n) * (PI_F / 128.0f);
    float cc = __cosf(theta), ss = __sinf(theta);
    v2f A;
    A.x = lo ? (xr * cc - xi * ss) : 0.0f;
    A.y = lo ? (xr * ss + xi * cc) : 0.0f;

    v8f acc = __builtin_amdgcn_wmma_f32_16x16x4_f32(
        false, A, false, B[0], (short)0, C[0], false, false);
#pragma unroll
    for (int ct = 1; ct < 4; ++ct) {
      v8f d = __builtin_amdgcn_wmma_f32_16x16x4_f32(
          false, A, false, B[ct], (short)0, C[ct], false, false);
#pragma unroll
      for (int r = 0; r < 8; ++r) acc[r] = fminf(acc[r], d[r]);
    }

#pragma unroll
    for (int r = 0; r < 8; ++r) {
      float v = acc[r];
      v = fminf(v, __shfl_xor(v, 1, 32));
      v = fminf(v, __shfl_xor(v, 2, 32));
      v = fminf(v, __shfl_xor(v, 4, 32));
      v = fminf(v, __shfl_xor(v, 8, 32));
      acc[r] = v;
    }
    if (lane == 0 || lane == 16) {
      float* dptr = dist + (size_t)l * MTA + t * 16 + (lane >> 1);
#pragma unroll
      for (int r = 0; r < 8; ++r) dptr[r] = acc[r] + px;
    }
  }
}

// ---------------------------------------------------------------------------
// Kernel 2: 256-thread block covers 64 consecutive l.  Stage the 192x64 dist
// slab into LDS once (async-to-LDS B128 when available), then 129-tap conv,
// row-normalize, softmin, circular mean of 4*theta -> ang[l].
// ---------------------------------------------------------------------------
__global__ __launch_bounds__(256) void bps_conv_softmax_kernel(
    const float* __restrict__ dist, const float* __restrict__ w_g,
    const float* __restrict__ temp_g, float* __restrict__ ang)
{
  __shared__ float tile[(TLB + 2 * NFW) * MTA];   // 192*64 floats = 48 KB
  __shared__ float red[256];
  __shared__ float wsh[FLEN];

  const int tid = threadIdx.x;
  const int l0  = blockIdx.x * TLB;

  if (tid < FLEN) wsh[tid] = w_g[tid];

  // ---- stage dist rows [l0-64, l0+127] into LDS, zero-padding OOB rows ----
#pragma unroll
  for (int it = 0; it < ((TLB + 2 * NFW) * MTA) / (4 * 256); ++it) {  // 12
    int e = it * 256 + tid;          // float4 index
    int j = e >> 4;                  // tile row (16 float4 per row)
    int c = (e & 15) << 2;           // column
    int g = l0 - NFW + j;            // global time index
    float* lp = &tile[j * MTA + c];
    if (g >= 0 && g < L_SYM) {
#ifdef HAVE_ASYNC_LDS
      __builtin_amdgcn_global_load_async_to_lds_b128(
          (gv4i*)(uintptr_t)(dist + (size_t)g * MTA + c),
          (lv4i*)(uint32_t)(uintptr_t)lp,
          0, 0);
#else
      *(float4*)lp = *(const float4*)(dist + (size_t)g * MTA + c);
#endif
    } else {
      float4 z; z.x = z.y = z.z = z.w = 0.0f;
      *(float4*)lp = z;
    }
  }
#ifdef HAVE_ASYNC_LDS
#if __has_builtin(__builtin_amdgcn_s_wait_asynccnt)
  __builtin_amdgcn_s_wait_asynccnt(0);
#else
  asm volatile("s_wait_asynccnt 0x0" ::: "memory");
#endif
#endif
  __syncthreads();

  const int m  = tid & 63;     // angle index
  const int lg = tid >> 6;     // 4 l's processed per pass

  float wsum = 0.0f;
  for (int k = 0; k < FLEN; ++k) wsum += wsh[k];
  const float invT = 1.0f / temp_g[0];
  const float th4  = 4.0f * (-0.25f * PI_F + (float)m * (PI_F / 128.0f));
  const float s4 = __sinf(th4), c4 = __cosf(th4);

  for (int p = 0; p < TLB / 4; ++p) {
    const int ll = p * 4 + lg;

    float acc = 0.0f;
    for (int k = 0; k < FLEN; ++k)
      acc += wsh[k] * tile[(ll + k) * MTA + m];
    float mvg = acc / wsum;

    // row sum over the 64 angles (4 independent 64-thread groups)
    red[tid] = mvg; __syncthreads();
    for (int s = 32; s > 0; s >>= 1) { if (m < s) red[tid] += red[tid + s]; __syncthreads(); }
    float rowsum = red[lg * 64]; __syncthreads();

    float z = -(mvg / rowsum) * invT;

    red[tid] = z; __syncthreads();
    for (int s = 32; s > 0; s >>= 1) { if (m < s) red[tid] = fmaxf(red[tid], red[tid + s]); __syncthreads(); }
    float zmax = red[lg * 64]; __syncthreads();

    float e = __expf(z - zmax);
    red[tid] = e; __syncthreads();
    for (int s = 32; s > 0; s >>= 1) { if (m < s) red[tid] += red[tid + s]; __syncthreads(); }
    float esum = red[lg * 64]; __syncthreads();
    float pr = e / esum;

    red[tid] = pr * s4; __syncthreads();
    for (int s = 32; s > 0; s >>= 1) { if (m < s) red[tid] += red[tid + s]; __syncthreads(); }
    float sdot = red[lg * 64]; __syncthreads();

    red[tid] = pr * c4; __syncthreads();
    for (int s = 32; s > 0; s >>= 1) { if (m < s) red[tid] += red[tid + s]; __syncthreads(); }
    float cdot = red[lg * 64];

    if (m == 0) ang[l0 + ll] = atan2f(sdot, cdot);
    __syncthreads();
  }
}

// ---------------------------------------------------------------------------
// Kernel 3: parallel unwrap (prefix scan of +-2pi jump corrections) and final
// rotation. Single 1024-thread block; wave shuffle scan + cross-wave LDS scan
// + serial chunk carry.  Output: [out.re,out.im] interleaved, then ph.
// ---------------------------------------------------------------------------
__global__ __launch_bounds__(1024) void bps_unwrap_out_kernel(
    const float* __restrict__ ang, const float* __restrict__ xr_g,
    const float* __restrict__ xi_g, float* __restrict__ out)
{
  __shared__ float wsum_sh[32];
  __shared__ float carry_sh;
  const int tid  = threadIdx.x;
  const int lane = tid & 31;
  const int wid  = tid >> 5;
  if (tid == 0) carry_sh = 0.0f;
  __syncthreads();

  for (int chunk = 0; chunk < L_SYM / 1024; ++chunk) {
    int idx = chunk * 1024 + tid;
    float a = ang[idx];

    float corr = 0.0f;
    if (idx > 0) {
      float dd = a - ang[idx - 1];
      float ddmod = fmodf(dd + PI_F, 2.0f * PI_F);
      if (ddmod < 0.0f) ddmod += 2.0f * PI_F;
      ddmod -= PI_F;
      if (ddmod == -PI_F && dd > 0.0f) ddmod = PI_F;
      float c = ddmod - dd;
      corr = (fabsf(dd) >= PI_F) ? c : 0.0f;
    }

    float v = corr;
    for (int off = 1; off < 32; off <<= 1) {
      float t = __shfl_up(v, off, 32);
      if (lane >= off) v += t;
    }
    if (lane == 31) wsum_sh[wid] = v;
    __syncthreads();

    float carry = carry_sh;
    float woff = 0.0f, total = 0.0f;
    for (int w = 0; w < 32; ++w) {
      float ws = wsum_sh[w];
      if (w < wid) woff += ws;
      total += ws;
    }

    float ph = (a + carry + woff + v) * 0.25f;
    float c = __cosf(ph), s = __sinf(ph);
    float xr = xr_g[idx], xi = xi_g[idx];
    out[2 * idx]         = xr * c - xi * s;
    out[2 * idx + 1]     = xr * s + xi * c;
    out[2 * L_SYM + idx] = ph;

    __syncthreads();
    if (tid == 0) carry_sh = carry + total;
    __syncthreads();
  }
}

// ---------------------------------------------------------------------------
extern "C" void kernel_launch(void* const* d_in, const int* in_sizes, int n_in,
                              void* d_out, int out_size, void* d_ws, size_t ws_size,
                              hipStream_t stream) {
  const float* xr   = (const float*)d_in[0];   // x_real      [L]
  const float* xi   = (const float*)d_in[1];   // x_imag      [L]
  const float* sr   = (const float*)d_in[2];   // symbols_real[64]
  const float* si   = (const float*)d_in[3];   // symbols_imag[64]
  const float* wf   = (const float*)d_in[4];   // avg_filter  [129]
  const float* temp = (const float*)d_in[5];   // temperature [1]
  float* out = (float*)d_out;

  float* dist = (float*)d_ws;                      // L*M floats = 16 MB
  float* ang  = dist + (size_t)L_SYM * MTA;        // L floats

  bps_dist_kernel<<<L_SYM / 8, 256, 0, stream>>>(xr, xi, sr, si, dist);
  bps_conv_softmax_kernel<<<L_SYM / TLB, 256, 0, stream>>>(dist, wf, temp, ang);
  bps_unwrap_out_kernel<<<1, 1024, 0, stream>>>(ang, xr, xi, out);
}